// MonoSIMoptimizerTorch_67525475828370
// MI455X (gfx1250) — compile-verified
//
#include <hip/hip_runtime.h>

// Complex cascade: G <- diag(exp(i*xi_l)) * (W_l @ G), l = 0..255, N = 512.
// fp32 WMMA (v_wmma_f32_16x16x4_f32) tiled GEMM per layer.
// CDNA5 paths: async global->LDS staging (ASYNCcnt) with double-buffered LDS,
// single barrier per K-tile, ping-pong G buffers in d_ws.

typedef __attribute__((ext_vector_type(2))) float v2f;
typedef __attribute__((ext_vector_type(8))) float v8f;

#define NDIM 512
#define LLAYERS 256
#define BM 64
#define BN 32
#define BK 32
#define KTILES (NDIM / BK)

// Async copy of 16 bytes: global -> LDS, tracked by ASYNCcnt (no VGPR staging).
__device__ __forceinline__ void async_copy16(unsigned lds_byte_off, const float* gsrc) {
    asm volatile("global_load_async_to_lds_b128 %0, %1, off"
                 :
                 : "v"(lds_byte_off), "v"((unsigned long long)gsrc)
                 : "memory");
}

__device__ __forceinline__ void wait_async0() {
    asm volatile("s_wait_asynccnt 0x0" ::: "memory");
}

__device__ __forceinline__ unsigned lds_off(const void* p) {
    // Generic LDS address keeps the wave-relative byte offset in addr[31:0].
    return (unsigned)(unsigned long long)p;
}

__global__ __launch_bounds__(256) void cascade_init(float* __restrict__ Gr,
                                                    float* __restrict__ Gi) {
    int idx = blockIdx.x * blockDim.x + threadIdx.x;
    if (idx < NDIM * NDIM) {
        int m = idx / NDIM;
        int n = idx - m * NDIM;
        Gr[idx] = (m == n) ? 1.0f : 0.0f;
        Gi[idx] = 0.0f;
    }
}

__global__ __launch_bounds__(256) void cascade_layer(
    const float* __restrict__ Wr, const float* __restrict__ Wi,
    const float* __restrict__ xi,
    const float* __restrict__ Gr_in, const float* __restrict__ Gi_in,
    float* __restrict__ Gr_out, float* __restrict__ Gi_out)
{
    __shared__ __align__(16) float sWr[2][BM][BK];
    __shared__ __align__(16) float sWi[2][BM][BK];
    __shared__ __align__(16) float sGr[2][BK][BN];
    __shared__ __align__(16) float sGi[2][BK][BN];

    const int tid  = threadIdx.x;
    const int lane = tid & 31;
    const int wave = tid >> 5;      // 0..7
    const int tm   = wave & 3;      // m-subtile within 64-row tile
    const int tn   = wave >> 2;     // n-subtile within 32-col tile

    const int blockM = blockIdx.y * BM;
    const int blockN = blockIdx.x * BN;

    // Issue all async copies for one K-tile into LDS buffer b.
    auto stage = [&](int b, int k0) {
#pragma unroll
        for (int i = 0; i < 2; ++i) {
            int f4  = tid + i * 256;          // 0..511 float4 slots in 64x32 tile
            int row = f4 >> 3;                // 8 float4 per 32-wide row
            int col = (f4 & 7) << 2;
            async_copy16(lds_off(&sWr[b][row][col]),
                         &Wr[(size_t)(blockM + row) * NDIM + k0 + col]);
            async_copy16(lds_off(&sWi[b][row][col]),
                         &Wi[(size_t)(blockM + row) * NDIM + k0 + col]);
        }
        {
            int row = tid >> 3;               // 32x32 tile: 1 float4 per thread
            int col = (tid & 7) << 2;
            async_copy16(lds_off(&sGr[b][row][col]),
                         &Gr_in[(size_t)(k0 + row) * NDIM + blockN + col]);
            async_copy16(lds_off(&sGi[b][row][col]),
                         &Gi_in[(size_t)(k0 + row) * NDIM + blockN + col]);
        }
    };

    v8f accP = {};  // Wr@Gr
    v8f accQ = {};  // Wi@Gi
    v8f accR = {};  // Wr@Gi
    v8f accS = {};  // Wi@Gr

    // Fragment addressing (per the CDNA5 f32 WMMA VGPR layouts):
    // A (16x4): lanes 0-15 hold K=k,k+1 ; lanes 16-31 hold K=k+2,k+3
    // B (4x16): symmetric, N across lanes within each half-wave
    const int aRow  = tm * 16 + (lane & 15);
    const int bCol  = tn * 16 + (lane & 15);
    const int kHalf = (lane >> 4) << 1;

    stage(0, 0);                               // prologue prefetch

    for (int t = 0; t < KTILES; ++t) {
        const int b = t & 1;
        wait_async0();                         // own async copies landed in LDS
        __syncthreads();                       // everyone's copies visible;
                                               // also proves all reads of the
                                               // other buffer (iter t-1) done
        if (t + 1 < KTILES) stage(b ^ 1, (t + 1) * BK);

#pragma unroll
        for (int kk = 0; kk < BK; kk += 4) {
            const int ka = kk + kHalf;
            v2f ar = { sWr[b][aRow][ka], sWr[b][aRow][ka + 1] };
            v2f ai = { sWi[b][aRow][ka], sWi[b][aRow][ka + 1] };
            v2f br = { sGr[b][ka][bCol], sGr[b][ka + 1][bCol] };
            v2f bi = { sGi[b][ka][bCol], sGi[b][ka + 1][bCol] };
            accP = __builtin_amdgcn_wmma_f32_16x16x4_f32(false, ar, false, br,
                                                         (short)0, accP, false, false);
            accQ = __builtin_amdgcn_wmma_f32_16x16x4_f32(false, ai, false, bi,
                                                         (short)0, accQ, false, false);
            accR = __builtin_amdgcn_wmma_f32_16x16x4_f32(false, ar, false, bi,
                                                         (short)0, accR, false, false);
            accS = __builtin_amdgcn_wmma_f32_16x16x4_f32(false, ai, false, br,
                                                         (short)0, accS, false, false);
        }
    }

    // ---- phase epilogue: out = e^{i*xi[m]} * (Cr + i*Ci) ----
    // C/D layout: lane -> N (lane&15), VGPR v -> M = v + 8*(lane>>4)
    const int nOut  = blockN + tn * 16 + (lane & 15);
    const int mBase = blockM + tm * 16 + ((lane >> 4) << 3);
#pragma unroll
    for (int v = 0; v < 8; ++v) {
        const int m = mBase + v;
        const float cr = accP[v] - accQ[v];
        const float ci = accR[v] + accS[v];
        float sn, cs;
        __sincosf(xi[m], &sn, &cs);
        Gr_out[(size_t)m * NDIM + nOut] = cs * cr - sn * ci;
        Gi_out[(size_t)m * NDIM + nOut] = cs * ci + sn * cr;
    }
}

__global__ __launch_bounds__(256) void cascade_pack(const float* __restrict__ Gr,
                                                    const float* __restrict__ Gi,
                                                    float* __restrict__ out) {
    int idx = blockIdx.x * blockDim.x + threadIdx.x;
    if (idx < NDIM * NDIM) {
        out[2 * idx + 0] = Gr[idx];
        out[2 * idx + 1] = Gi[idx];
    }
}

extern "C" void kernel_launch(void* const* d_in, const int* in_sizes, int n_in,
                              void* d_out, int out_size, void* d_ws, size_t ws_size,
                              hipStream_t stream) {
    const float* W_real = (const float*)d_in[0];   // [L, N, N]
    const float* W_imag = (const float*)d_in[1];   // [L, N, N]
    const float* xi     = (const float*)d_in[2];   // [L, N]

    const size_t mat = (size_t)NDIM * NDIM;
    float* Gr0 = (float*)d_ws;
    float* Gi0 = Gr0 + mat;
    float* Gr1 = Gi0 + mat;
    float* Gi1 = Gr1 + mat;

    const int threads = 256;
    const int blocksFlat = (int)((mat + threads - 1) / threads);

    cascade_init<<<blocksFlat, threads, 0, stream>>>(Gr0, Gi0);

    dim3 grid(NDIM / BN, NDIM / BM);   // (16, 8) = 128 workgroups
    for (int l = 0; l < LLAYERS; ++l) {
        const float* srcR = (l & 1) ? Gr1 : Gr0;
        const float* srcI = (l & 1) ? Gi1 : Gi0;
        float* dstR = (l & 1) ? Gr0 : Gr1;
        float* dstI = (l & 1) ? Gi0 : Gi1;
        cascade_layer<<<grid, threads, 0, stream>>>(
            W_real + (size_t)l * mat, W_imag + (size_t)l * mat,
            xi + (size_t)l * NDIM, srcR, srcI, dstR, dstI);
    }

    // LLAYERS is even -> final G lives in buffer 0
    cascade_pack<<<blocksFlat, threads, 0, stream>>>(Gr0, Gi0, (float*)d_out);
}